// HashEncoding_44822278701158
// MI455X (gfx1250) — compile-verified
//
#include <hip/hip_runtime.h>
#include <stdint.h>

// ---------------------------------------------------------------------------
// Instant-NGP hash encoding, CDNA5 (gfx1250, wave32).
// Thread t -> point n = t>>4, level lvl = t&15.
// Output float2 per (point, level): out[n*16 + lvl].
//
// Memory strategy (MI455X): 42 MB embedding table is L2-resident (192 MB L2);
// ~2 GB of random float2 gathers hit L2. Coord stream (24 MB, read once) and
// output stream (256 MB, written once) use non-temporal hints so they do not
// evict the table. HBM traffic ~320 MB -> ~14 us floor at 23.3 TB/s.
// ---------------------------------------------------------------------------

#define HE_NUM_LEVELS 16
#define HE_T_MASK     524287u   // 2^19 - 1 : all hashed levels have exactly 2^19 entries
#define HE_PI1        2654435761u
#define HE_PI2        805459861u

typedef float v2f __attribute__((ext_vector_type(2)));

// [0..15] = per-level resolution, [16..31] = per-level table offset (entries)
__device__ const int g_he_tab[32] = {
    16, 20, 25, 32, 40, 50, 64, 80, 101, 128, 161, 203, 256, 322, 406, 512,
    0, 4913, 14174, 31750, 67687, 136608, 269259, 543884,
    1068172, 1592460, 2116748, 2641036, 3165324, 3689612, 4213900, 4738188
};

__global__ void __launch_bounds__(256)
HashEncoding_44822278701158_kernel(const float*  __restrict__ coords,
                                   const float2* __restrict__ emb,
                                   v2f*          __restrict__ out,
                                   int npoints)
{
    __shared__ int s_tab[32];

    // ---- stage the 128-byte level table into LDS (async path on CDNA5) ----
    if (threadIdx.x < 32) {
#if __has_builtin(__builtin_amdgcn_global_load_async_to_lds_b32)
        __builtin_amdgcn_global_load_async_to_lds_b32(
            (__attribute__((address_space(1))) int*)&g_he_tab[threadIdx.x],
            (__attribute__((address_space(3))) int*)&s_tab[threadIdx.x],
            /*offset=*/0, /*cpol=*/0);
#else
        s_tab[threadIdx.x] = g_he_tab[threadIdx.x];
#endif
    }
#if __has_builtin(__builtin_amdgcn_global_load_async_to_lds_b32)
#if __has_builtin(__builtin_amdgcn_s_wait_asynccnt)
    __builtin_amdgcn_s_wait_asynccnt(0);
#else
    asm volatile("s_wait_asynccnt 0x0" ::: "memory");
#endif
#endif
    __syncthreads();

    const long long t = (long long)blockIdx.x * blockDim.x + threadIdx.x;
    if (t >= (long long)npoints * HE_NUM_LEVELS) return;

    const int n   = (int)(t >> 4);
    const int lvl = (int)(t & 15);

    const int ires = s_tab[lvl];
    const int off  = s_tab[16 + lvl];
    const bool hashed = (lvl >= 7);

    // speculative read-ahead of the coord stream (global_prefetch_b8)
    __builtin_prefetch(&coords[3 * (n + 4096)], 0, 1);

    // coords: streamed once -> non-temporal loads (TH_LOAD_NT), keep L2 for the table
    const float px = __builtin_nontemporal_load(&coords[3 * n + 0]);
    const float py = __builtin_nontemporal_load(&coords[3 * n + 1]);
    const float pz = __builtin_nontemporal_load(&coords[3 * n + 2]);

    const float fres = (float)ires;
    const float sx = px * fres, sy = py * fres, sz = pz * fres;
    const float bx = floorf(sx), by = floorf(sy), bz = floorf(sz);
    const float fx = sx - bx,   fy = sy - by,   fz = sz - bz;

    const int ix = (int)bx, iy = (int)by, iz = (int)bz;
    const int x0 = min(max(ix,     0), ires), x1 = min(max(ix + 1, 0), ires);
    const int y0 = min(max(iy,     0), ires), y1 = min(max(iy + 1, 0), ires);
    const int z0 = min(max(iz,     0), ires), z1 = min(max(iz + 1, 0), ires);

    // hashed partial products (PI0 == 1, so x contributes raw)
    const uint32_t hy0 = (uint32_t)y0 * HE_PI1, hy1 = (uint32_t)y1 * HE_PI1;
    const uint32_t hz0 = (uint32_t)z0 * HE_PI2, hz1 = (uint32_t)z1 * HE_PI2;

    // dense partial products
    const int d1  = ires + 1;
    const int dy0 = y0 * d1,      dy1 = y1 * d1;
    const int dz0 = z0 * d1 * d1, dz1 = z1 * d1 * d1;

    // branchless hashed/dense index select (v_cndmask), 8 corners
    const int i000 = hashed ? (int)(((uint32_t)x0 ^ hy0 ^ hz0) & HE_T_MASK) : (x0 + dy0 + dz0);
    const int i100 = hashed ? (int)(((uint32_t)x1 ^ hy0 ^ hz0) & HE_T_MASK) : (x1 + dy0 + dz0);
    const int i010 = hashed ? (int)(((uint32_t)x0 ^ hy1 ^ hz0) & HE_T_MASK) : (x0 + dy1 + dz0);
    const int i110 = hashed ? (int)(((uint32_t)x1 ^ hy1 ^ hz0) & HE_T_MASK) : (x1 + dy1 + dz0);
    const int i001 = hashed ? (int)(((uint32_t)x0 ^ hy0 ^ hz1) & HE_T_MASK) : (x0 + dy0 + dz1);
    const int i101 = hashed ? (int)(((uint32_t)x1 ^ hy0 ^ hz1) & HE_T_MASK) : (x1 + dy0 + dz1);
    const int i011 = hashed ? (int)(((uint32_t)x0 ^ hy1 ^ hz1) & HE_T_MASK) : (x0 + dy1 + dz1);
    const int i111 = hashed ? (int)(((uint32_t)x1 ^ hy1 ^ hz1) & HE_T_MASK) : (x1 + dy1 + dz1);

    // issue all 8 gathers up front (global_load_b64, default RT policy: cache in L2)
    const float2* __restrict__ tab = emb + off;
    const float2 e000 = tab[i000];
    const float2 e100 = tab[i100];
    const float2 e010 = tab[i010];
    const float2 e110 = tab[i110];
    const float2 e001 = tab[i001];
    const float2 e101 = tab[i101];
    const float2 e011 = tab[i011];
    const float2 e111 = tab[i111];

    const float ux = 1.0f - fx, uy = 1.0f - fy, uz = 1.0f - fz;

    float accx = 0.0f, accy = 0.0f, w;
    w = ux * uy * uz; accx += w * e000.x; accy += w * e000.y;
    w = fx * uy * uz; accx += w * e100.x; accy += w * e100.y;
    w = ux * fy * uz; accx += w * e010.x; accy += w * e010.y;
    w = fx * fy * uz; accx += w * e110.x; accy += w * e110.y;
    w = ux * uy * fz; accx += w * e001.x; accy += w * e001.y;
    w = fx * uy * fz; accx += w * e101.x; accy += w * e101.y;
    w = ux * fy * fz; accx += w * e011.x; accy += w * e011.y;
    w = fx * fy * fz; accx += w * e111.x; accy += w * e111.y;

    // consecutive lanes (levels) -> contiguous 256B per wave: fully coalesced.
    // Output is written once and never re-read: non-temporal store (TH_STORE_NT)
    // so the 256 MB stream does not evict the gather table from L2.
    v2f r;
    r.x = accx;
    r.y = accy;
    __builtin_nontemporal_store(r, &out[(long long)n * HE_NUM_LEVELS + lvl]);
}

extern "C" void kernel_launch(void* const* d_in, const int* in_sizes, int n_in,
                              void* d_out, int out_size, void* d_ws, size_t ws_size,
                              hipStream_t stream)
{
    const float*  coords = (const float*)d_in[0];    // [N, 3] float32
    const float2* emb    = (const float2*)d_in[1];   // [TOTAL_ENTRIES, 2] float32
    v2f*          out    = (v2f*)d_out;              // [N, 16*2] float32

    const int npoints = in_sizes[0] / 3;
    const long long total = (long long)npoints * HE_NUM_LEVELS;
    const int block = 256;
    const int grid  = (int)((total + block - 1) / block);

    HashEncoding_44822278701158_kernel<<<grid, block, 0, stream>>>(coords, emb, out, npoints);
}